// valueLayer_33483565040210
// MI455X (gfx1250) — compile-verified
//
#include <hip/hip_runtime.h>
#include <hip/hip_bf16.h>
#include <math.h>

// ---------------- problem constants ----------------
#define B_    32
#define CT    512
#define HW    64        // H*W = 8*8
#define EMB   4096
#define EACH  64
#define LDA   64
#define INP   256
#define ROWS  2048      // B_ * HW
#define BN_EPS   1e-5f
#define NORM_EPS 1e-12f

typedef __attribute__((ext_vector_type(2))) float v2f;
typedef __attribute__((ext_vector_type(4))) float v4f;
typedef __attribute__((ext_vector_type(8))) float v8f;
typedef __attribute__((ext_vector_type(4))) unsigned int u32x4;
typedef __attribute__((ext_vector_type(8))) int i32x8;
typedef __attribute__((ext_vector_type(4))) int i32x4;

#if defined(__has_builtin)
# if __has_builtin(__builtin_amdgcn_tensor_load_to_lds)
#  define HAVE_TDM 1
# endif
#endif

// ---------------- ws layout (float slots) ----------------
#define WS_ENC   0       // int[2048]
#define WS_LSE   2048    // float[2048]
#define WS_BNA   4096    // float[4096]
#define WS_BNB   8192    // float[4096]
#define WS_INVW  12288   // float[256]
#define WS_INVX  12544   // float[2048]

// ============================================================
// K1: LDA projection + argmin distance within label block -> enc
// ============================================================
__global__ __launch_bounds__(64) void k_enc(
    const float* __restrict__ ftin, const float* __restrict__ lda_w,
    const float* __restrict__ lda_b, const float* __restrict__ cc,
    const int* __restrict__ labels, int* __restrict__ enc)
{
    int r = blockIdx.x;             // row = b*64 + pos
    int b = r >> 6, pos = r & 63;
    int t = threadIdx.x;            // 0..63

    __shared__ float ft[LDA];
    float acc = lda_b[t];
    const float* fbase = ftin + (size_t)b * CT * HW + pos;
    const float* wbase = lda_w + (size_t)t * CT;
    for (int c = 0; c < CT; ++c)
        acc += fbase[(size_t)c * HW] * wbase[c];
    ft[t] = acc;
    __syncthreads();

    int blockbase = labels[b] * EACH;
    const float* ccrow = cc + (size_t)(blockbase + t) * LDA;
    float dot = 0.f, cn = 0.f;
    for (int k = 0; k < LDA; ++k) {
        float cv = ccrow[k];
        dot += ft[k] * cv;
        cn  += cv * cv;
    }
    // argmin d == argmax (2*dot - cn); ties -> earliest index
    float score = 2.f * dot - cn;

    __shared__ float sv[64];
    __shared__ int   si[64];
    sv[t] = score; si[t] = t;
    __syncthreads();
    for (int s = 32; s > 0; s >>= 1) {
        if (t < s) {
            if (sv[t + s] > sv[t]) { sv[t] = sv[t + s]; si[t] = si[t + s]; }
        }
        __syncthreads();
    }
    if (t == 0) enc[r] = blockbase + si[0];
}

// ============================================================
// K2: per-row logsumexp of scores
// ============================================================
__global__ __launch_bounds__(256) void k_lse(
    const float* __restrict__ scores, float* __restrict__ lse)
{
    int r = blockIdx.x; int b = r >> 6, pos = r & 63;
    int t = threadIdx.x;
    const float* srow = scores + (size_t)b * EMB * HW + pos;

    __shared__ float red[256];
    float m = -INFINITY;
    for (int c = t; c < EMB; c += 256) m = fmaxf(m, srow[(size_t)c * HW]);
    red[t] = m; __syncthreads();
    for (int s = 128; s > 0; s >>= 1) {
        if (t < s) red[t] = fmaxf(red[t], red[t + s]);
        __syncthreads();
    }
    m = red[0]; __syncthreads();

    float se = 0.f;
    for (int c = t; c < EMB; c += 256) se += expf(srow[(size_t)c * HW] - m);
    red[t] = se; __syncthreads();
    for (int s = 128; s > 0; s >>= 1) {
        if (t < s) red[t] += red[t + s];
        __syncthreads();
    }
    if (t == 0) lse[r] = m + logf(red[0]);
}

// ============================================================
// K3: BN stats per channel -> fused affine bnA, bnB
// ============================================================
__global__ __launch_bounds__(256) void k_bn(
    const float* __restrict__ scores, const float* __restrict__ gamma,
    const float* __restrict__ beta, float* __restrict__ bnA, float* __restrict__ bnB)
{
    int c = blockIdx.x, t = threadIdx.x;
    float s = 0.f, sq = 0.f;
    for (int i = t; i < ROWS; i += 256) {
        int b = i >> 6, pos = i & 63;
        float v = scores[((size_t)b * EMB + c) * HW + pos];
        s += v; sq += v * v;
    }
    __shared__ float r1[256], r2[256];
    r1[t] = s; r2[t] = sq; __syncthreads();
    for (int st = 128; st > 0; st >>= 1) {
        if (t < st) { r1[t] += r1[t + st]; r2[t] += r2[t + st]; }
        __syncthreads();
    }
    if (t == 0) {
        float mean = r1[0] * (1.0f / ROWS);
        float var  = r2[0] * (1.0f / ROWS) - mean * mean;
        float rstd = rsqrtf(var + BN_EPS);
        float g = gamma[c];
        bnA[c] = rstd * g;
        bnB[c] = beta[c] - mean * rstd * g;
    }
}

// ============================================================
// K4: conv_w per-row inverse norm
// ============================================================
__global__ __launch_bounds__(256) void k_wnorm(
    const float* __restrict__ w, float* __restrict__ invw)
{
    int o = blockIdx.x, t = threadIdx.x;
    const float* row = w + (size_t)o * EMB;
    float sq = 0.f;
    for (int k = t; k < EMB; k += 256) { float v = row[k]; sq += v * v; }
    __shared__ float red[256];
    red[t] = sq; __syncthreads();
    for (int s = 128; s > 0; s >>= 1) {
        if (t < s) red[t] += red[t + s];
        __syncthreads();
    }
    if (t == 0) invw[o] = 1.0f / fmaxf(sqrtf(red[0]), NORM_EPS);
}

// ============================================================
// K5: per-row inverse norm of relu(BN(scores))
// ============================================================
__global__ __launch_bounds__(256) void k_xnorm(
    const float* __restrict__ scores, const float* __restrict__ bnA,
    const float* __restrict__ bnB, float* __restrict__ invx)
{
    int r = blockIdx.x; int b = r >> 6, pos = r & 63;
    int t = threadIdx.x;
    const float* srow = scores + (size_t)b * EMB * HW + pos;
    float sq = 0.f;
    for (int c = t; c < EMB; c += 256) {
        float v = fmaxf(srow[(size_t)c * HW] * bnA[c] + bnB[c], 0.f);
        sq += v * v;
    }
    __shared__ float red[256];
    red[t] = sq; __syncthreads();
    for (int s = 128; s > 0; s >>= 1) {
        if (t < s) red[t] += red[t + s];
        __syncthreads();
    }
    if (t == 0) invx[r] = 1.0f / fmaxf(sqrtf(red[0]), NORM_EPS);
}

// ============================================================
// K6: zero the loss accumulator
// ============================================================
__global__ void k_zero(float* loss) { if (threadIdx.x == 0) *loss = 0.f; }

// ============================================================
// K7: KD loss
// ============================================================
__global__ __launch_bounds__(256) void k_loss(
    const float* __restrict__ scores, const float* __restrict__ ts,
    const int* __restrict__ enc, const float* __restrict__ lse,
    float* __restrict__ loss)
{
    int r = blockIdx.x; int b = r >> 6, pos = r & 63;
    int t = threadIdx.x;
    const float* srow = scores + (size_t)b * EMB * HW + pos;
    const float* trow = ts + (size_t)enc[r] * EMB;
    float L = lse[r];
    float acc = 0.f;
    for (int c = t; c < EMB; c += 256) {
        float tv = trow[c];
        float logp = srow[(size_t)c * HW] - L;
        float tl = (tv > 0.f) ? tv * logf(tv) : 0.f;
        acc += tl - tv * logp;
    }
    __shared__ float red[256];
    red[t] = acc; __syncthreads();
    for (int s = 128; s > 0; s >>= 1) {
        if (t < s) red[t] += red[t + s];
        __syncthreads();
    }
    if (t == 0) atomicAdd(loss, red[0] * (1.0f / ROWS));
}

// ============================================================
// K8: double-buffered, TDM-fed, register-blocked f32 WMMA GEMM
//   C[2048x256] = X[2048x4096] * W^T
//   - Block (256 thr, 8 waves) computes 64 rows x 128 cols;
//     wave = 2x2 tiles of 16x16 (4 v8f accumulators).
//   - K chunked by 32; LDS layout [row][36] (stride-36 pad) gives
//     conflict-free single ds_load_b64 fragments.
//   - B tile (32x128 of conv_w) DMA'd by the Tensor Data Mover
//     (pad_interval=32 dw, pad_amount=4 dw -> stride-36 layout);
//     issued by ONE wave via a scalar (readfirstlane) branch.
//   - Double-buffered: DMA + A staging of chunk t+1 overlap the
//     WMMAs of chunk t; s_wait_tensorcnt sits after the compute.
//   - Epilogue folds scale * invx[row] * invw[col].
// ============================================================
#define KC 32
#define BSTRIDE 36      // floats per LDS row (32 data + 4 pad)
#define ASZ (64 * BSTRIDE)
#define BSZ (128 * BSTRIDE)

#ifdef HAVE_TDM
// issue one 2D TDM load: 32(K) x 128(O) tile of conv_w -> padded LDS
__device__ __forceinline__ void tdm_load_b(const float* gtile, unsigned ldsOff)
{
    unsigned long long ga = (unsigned long long)(uintptr_t)gtile;
    u32x4 g0;
    g0.x = 1u;                                      // count=1, user mode
    g0.y = ldsOff;                                  // lds_addr (bytes)
    g0.z = (unsigned)ga;                            // global_addr[31:0]
    g0.w = (unsigned)((ga >> 32) & 0x1FFFFFFull)    // global_addr[56:32]
         | (2u << 30);                              // type=2 ("image")
    i32x8 g1 = {};
    g1.s0 = (int)((2u << 16)      // data_size = 4B
                | (1u << 20)      // pad_enable
                | (4u << 22)      // pad_interval: 32 dwords
                | (3u << 25));    // pad_amount:   4 dwords
    g1.s1 = (int)((EMB & 0xFFFF) << 16);   // tensor_dim0 = 4096 (bits 79:48)
    g1.s2 = (int)((INP & 0xFFFF) << 16);   // tensor_dim1 = 256  (bits 111:80)
    g1.s3 = (int)(KC << 16);               // tile_dim0 = 32     (bits 127:112)
    g1.s4 = 128;                           // tile_dim1 = 128    (bits 143:128)
    g1.s5 = EMB;                           // tensor_dim0_stride = 4096 (bits 207:160)
    i32x4 z4 = {0, 0, 0, 0};
#if defined(__clang_major__) && (__clang_major__ >= 23)
    i32x8 z8 = {};
    __builtin_amdgcn_tensor_load_to_lds(g0, g1, z4, z4, z8, 0);
#else
    __builtin_amdgcn_tensor_load_to_lds(g0, g1, z4, z4, 0);
#endif
}
#endif

__global__ __launch_bounds__(256) void k_value(
    const float* __restrict__ scores, const float* __restrict__ conv_w,
    const float* __restrict__ in_block, const float* __restrict__ bnA,
    const float* __restrict__ bnB, const float* __restrict__ invx,
    const float* __restrict__ invw, const float* __restrict__ scale_p,
    float* __restrict__ features, float* __restrict__ value)
{
    int mblk = blockIdx.x >> 1;          // 0..31 == batch index b
    int nblk = blockIdx.x & 1;           // 0..1
    int mBase = mblk * 64;               // rows = pos 0..63 of batch mblk
    int nBase = nblk * 128;

    int tid  = threadIdx.x;
    int wave = tid >> 5;                 // 0..7
    int lane = tid & 31;
    int half = lane >> 4;                // 0: K={0,1}, 1: K={2,3}
    int l    = lane & 15;
    int wm   = wave >> 2;                // 0..1 : wave M group (32 rows)
    int wn   = wave & 3;                 // 0..3 : wave N group (32 cols)
    int ko   = half * 2;

    __shared__ float As[2][ASZ];         // 2 x  9.0 KB
    __shared__ float Bs[2][BSZ];         // 2 x 18.0 KB

    // staging mappings
    int aPos = tid & 63;                 // pos within batch (row)
    int aK0  = tid >> 6;                 // 0..3, step 4 -> covers 32 k
    const float* sbase = scores + (size_t)mblk * EMB * HW;   // + c*64 + pos
    const float* wrow  = conv_w + (size_t)nBase * EMB;

    // per-lane fragment offsets within a buffer (conflict-free stride-36)
    int oA0 = (wm * 32      + l) * BSTRIDE + ko;
    int oA1 = (wm * 32 + 16 + l) * BSTRIDE + ko;
    int oB0 = (wn * 32      + l) * BSTRIDE + ko;
    int oB1 = (wn * 32 + 16 + l) * BSTRIDE + ko;

    // scalar predicate: exactly one wave (wave 0) owns the TDM issue
    bool tdmWave = (__builtin_amdgcn_readfirstlane(tid) < 32);
    (void)tdmWave;
#ifdef HAVE_TDM
    unsigned ldsB0 = (unsigned)(uintptr_t)(void*)&Bs[0][0]; // flat low 32 = LDS offset
    unsigned ldsB1 = (unsigned)(uintptr_t)(void*)&Bs[1][0];
#endif

    v8f acc[2][2] = {};

    // ---- staging helpers ----
    auto stageA = [&](int kc, float* Ab) {
#pragma unroll
        for (int i = 0; i < 8; ++i) {
            int kk = aK0 + i * 4;              // 0..31
            int c  = kc + kk;
            float v = sbase[(size_t)c * HW + aPos];
            Ab[aPos * BSTRIDE + kk] = fmaxf(v * bnA[c] + bnB[c], 0.f);
        }
    };
#ifndef HAVE_TDM
    auto stageB = [&](int kc, float* Bb) {
#pragma unroll
        for (int i = 0; i < 4; ++i) {
            int idx = tid + i * 256;           // 0..1023
            int o   = idx >> 3;                // 0..127
            int cq  = idx & 7;                 // float4 within 32-k chunk
            v4f wv = *(const v4f*)(wrow + (size_t)o * EMB + kc + cq * 4);
            *(v4f*)&Bb[o * BSTRIDE + cq * 4] = wv;  // 16B-aligned (144B rows)
        }
    };
#endif
    auto compute = [&](const float* Ab, const float* Bb) {
#pragma unroll
        for (int k = 0; k < KC; k += 4) {
            v2f a0 = *(const v2f*)&Ab[oA0 + k];
            v2f a1 = *(const v2f*)&Ab[oA1 + k];
            v2f b0 = *(const v2f*)&Bb[oB0 + k];
            v2f b1 = *(const v2f*)&Bb[oB1 + k];
            acc[0][0] = __builtin_amdgcn_wmma_f32_16x16x4_f32(
                false, a0, false, b0, (short)0, acc[0][0], false, false);
            acc[0][1] = __builtin_amdgcn_wmma_f32_16x16x4_f32(
                false, a0, false, b1, (short)0, acc[0][1], false, false);
            acc[1][0] = __builtin_amdgcn_wmma_f32_16x16x4_f32(
                false, a1, false, b0, (short)0, acc[1][0], false, false);
            acc[1][1] = __builtin_amdgcn_wmma_f32_16x16x4_f32(
                false, a1, false, b1, (short)0, acc[1][1], false, false);
        }
    };

    // ---- prologue: stage chunk 0 into buffer 0 ----
#ifdef HAVE_TDM
    if (tdmWave) tdm_load_b(wrow, ldsB0);
#else
    stageB(0, Bs[0]);
#endif
    stageA(0, As[0]);
#ifdef HAVE_TDM
    if (tdmWave) __builtin_amdgcn_s_wait_tensorcnt(0);
#endif
    __syncthreads();

    // ---- main loop: unrolled x2 over the double buffer ----
    for (int kc = 0; kc < EMB; kc += 2 * KC) {
        // stage chunk kc+KC into buf1 while computing buf0
#ifdef HAVE_TDM
        if (tdmWave) tdm_load_b(wrow + kc + KC, ldsB1);
#else
        stageB(kc + KC, Bs[1]);
#endif
        stageA(kc + KC, As[1]);
        compute(As[0], Bs[0]);
#ifdef HAVE_TDM
        if (tdmWave) __builtin_amdgcn_s_wait_tensorcnt(0);
#endif
        __syncthreads();

        // stage chunk kc+2*KC into buf0 while computing buf1
        bool more = (kc + 2 * KC) < EMB;
        if (more) {
#ifdef HAVE_TDM
            if (tdmWave) tdm_load_b(wrow + kc + 2 * KC, ldsB0);
#else
            stageB(kc + 2 * KC, Bs[0]);
#endif
            stageA(kc + 2 * KC, As[0]);
        }
        compute(As[1], Bs[1]);
        if (more) {
#ifdef HAVE_TDM
            if (tdmWave) __builtin_amdgcn_s_wait_tensorcnt(0);
#endif
            __syncthreads();
        }
    }

    // ---- epilogue: value = scale*invx[m]*invw[n]*acc ; features = value + in_block
    float sc = scale_p[0];
#pragma unroll
    for (int j = 0; j < 2; ++j) {
        int n  = nBase + wn * 32 + j * 16 + l;
        float iw = invw[n] * sc;
#pragma unroll
        for (int i = 0; i < 2; ++i) {
            int mTile = mBase + wm * 32 + i * 16;
#pragma unroll
            for (int v = 0; v < 8; ++v) {
                int m = mTile + v + half * 8;        // D: M = v + half*8, N = l
                float val = acc[i][j][v] * iw * invx[m];
                int pos = m & 63;                    // b == mblk
                size_t idx = ((size_t)mblk * INP + n) * HW + pos;
                value[idx]    = val;
                features[idx] = val + in_block[idx]; // VALUE_RATIO = 1.0
            }
        }
    }
}

// ============================================================
extern "C" void kernel_launch(void* const* d_in, const int* in_sizes, int n_in,
                              void* d_out, int out_size, void* d_ws, size_t ws_size,
                              hipStream_t stream)
{
    const float* feature_teacher = (const float*)d_in[0];
    const float* scores          = (const float*)d_in[1];
    const float* in_block        = (const float*)d_in[2];
    const int*   labels          = (const int*)  d_in[3];
    const float* lda_w           = (const float*)d_in[4];
    const float* lda_b           = (const float*)d_in[5];
    const float* cluster_centers = (const float*)d_in[6];
    const float* teacher_scores  = (const float*)d_in[7];
    const float* bn_gamma        = (const float*)d_in[8];
    const float* bn_beta         = (const float*)d_in[9];
    const float* conv_w          = (const float*)d_in[10];
    const float* scale           = (const float*)d_in[11];

    float* ws  = (float*)d_ws;
    int*   enc = (int*)d_ws + WS_ENC;
    float* lse  = ws + WS_LSE;
    float* bnA  = ws + WS_BNA;
    float* bnB  = ws + WS_BNB;
    float* invw = ws + WS_INVW;
    float* invx = ws + WS_INVX;

    float* out      = (float*)d_out;
    float* features = out;                       // (B, INP, H, W)
    float* value    = out + (size_t)B_ * INP * HW;
    float* loss     = out + (size_t)2 * B_ * INP * HW;

    k_enc  <<<ROWS, 64,  0, stream>>>(feature_teacher, lda_w, lda_b,
                                      cluster_centers, labels, enc);
    k_lse  <<<ROWS, 256, 0, stream>>>(scores, lse);
    k_bn   <<<EMB,  256, 0, stream>>>(scores, bn_gamma, bn_beta, bnA, bnB);
    k_wnorm<<<INP,  256, 0, stream>>>(conv_w, invw);
    k_xnorm<<<ROWS, 256, 0, stream>>>(scores, bnA, bnB, invx);
    k_zero <<<1, 64, 0, stream>>>(loss);
    k_loss <<<ROWS, 256, 0, stream>>>(scores, teacher_scores, enc, lse, loss);
    k_value<<<64,   256, 0, stream>>>(scores, conv_w, in_block, bnA, bnB,
                                      invx, invw, scale, features, value);
}